// MultiheadAttention_19456201850889
// MI455X (gfx1250) — compile-verified
//
#include <hip/hip_runtime.h>

// MI455X / gfx1250 fused attention block, fp32 end-to-end via V_WMMA_F32_16X16X4_F32.
// Async global->LDS staging (GLOBAL_LOAD_ASYNC_TO_LDS_B128 / ASYNCcnt) with
// double-buffered LDS and 1-stage software pipeline (s_wait_asynccnt <= in-flight).
// B=8, S=2048, H=1024.

typedef __attribute__((ext_vector_type(2))) float v2f;
typedef __attribute__((ext_vector_type(8))) float v8f;

#define TILE_M 128
#define TILE_N 128
#define TILE_K 32
#define LSTR   36   // LDS row stride (floats): 144B rows -> 16B-aligned (async b128 OK),
                    // 36*m mod 64 distinct for m=0..15 -> conflict-free fragment reads

// Async copy of 16 bytes global -> LDS, tracked by ASYNCcnt.
__device__ __forceinline__ void async_copy_b128(const float* lds_dst, const float* gsrc)
{
    unsigned lds_off = (unsigned)(unsigned long long)lds_dst;   // low 32 bits = LDS offset
    asm volatile("global_load_async_to_lds_b128 %0, %1, off"
                 :: "v"(lds_off), "v"((unsigned long long)gsrc)
                 : "memory");
}
// Wait until at most N async ops remain outstanding (async loads retire in order).
template<int N>
__device__ __forceinline__ void wait_async_le()
{
    asm volatile("s_wait_asynccnt %0" :: "i"(N) : "memory");
}

// Epilogues: 0 = C+bias[n] ; 1 = C*scale + attn_bias, mask->-1e12 ; 2 = plain ; 3 = C+bias[n]+residual
template<int EPI, bool BT>
__global__ __launch_bounds__(256)
void gemm_wmma_f32(const float* __restrict__ A, const float* __restrict__ B,
                   float* __restrict__ C, int M, int N, int K,
                   int lda, int ldb, int ldc,
                   const float* __restrict__ bias,
                   const float* __restrict__ add2d,
                   const int* __restrict__ mask,
                   float scale,
                   long sAz, long sBz, long sCz, long sAddz)
{
    __shared__ float As[2][TILE_M * LSTR];
    __shared__ float Bs[2][TILE_N * LSTR];

    const int tid   = threadIdx.x;
    const int lane  = tid & 31;
    const int wave  = tid >> 5;
    const int waveM = wave & 3;    // 4 waves along M -> 32 rows each
    const int waveN = wave >> 2;   // 2 waves along N -> 64 cols each
    const int z  = blockIdx.z;
    const int m0 = blockIdx.y * TILE_M;
    const int n0 = blockIdx.x * TILE_N;

    const float* Abase = A + (long)z * sAz + (long)m0 * lda;
    const float* Bbase = BT ? (B + (long)z * sBz + (long)n0 * ldb)
                            : (B + (long)z * sBz + n0);

    v8f acc[2][4] = {};

    const int mlo   = lane & 15;
    const int khalf = (lane >> 4) << 1;   // lanes 0-15 -> K0,K1 ; lanes 16-31 -> K2,K3

    // Per-thread staging coordinates (row-major, K-contiguous tiles): 1024 b128 chunks.
    const int srow = tid >> 3;            // 0..31 (row within 128, covered in 4 passes)
    const int scol = (tid & 7) << 2;      // 0,4,..,28

    constexpr int NASY = BT ? 8 : 4;      // async ops issued per stage per thread

    // Stage tile starting at column k0 into LDS buffer `buf`.
    auto stage_tile = [&](int k0, int buf) {
        const float* Ag = Abase + k0;
        #pragma unroll
        for (int p = 0; p < 4; ++p) {
            const int row = srow + (p << 5);
            async_copy_b128(&As[buf][row * LSTR + scol], Ag + (long)row * lda + scol);
        }
        if constexpr (BT) {               // B is N x K row-major: direct async copy
            const float* Bg = Bbase + k0;
            #pragma unroll
            for (int p = 0; p < 4; ++p) {
                const int row = srow + (p << 5);
                async_copy_b128(&Bs[buf][row * LSTR + scol], Bg + (long)row * ldb + scol);
            }
        } else {                          // B is K x N row-major (V): transpose via VGPRs
            const float* Bg = Bbase + (long)k0 * ldb;
            __builtin_prefetch(Bg + (long)TILE_K * ldb, 0, 0);
            #pragma unroll
            for (int p = 0; p < 4; ++p) {
                int idx = tid + (p << 8);
                int kr  = idx >> 5;                 // 0..31
                int n   = (idx & 31) << 2;          // 0..124
                const float4 d = *(const float4*)(Bg + (long)kr * ldb + n);
                Bs[buf][(n + 0) * LSTR + kr] = d.x;
                Bs[buf][(n + 1) * LSTR + kr] = d.y;
                Bs[buf][(n + 2) * LSTR + kr] = d.z;
                Bs[buf][(n + 3) * LSTR + kr] = d.w;
            }
        }
    };

    const int T = K / TILE_K;
    stage_tile(0, 0);                     // prologue: tile 0 -> buffer 0

    for (int t = 0; t < T; ++t) {
        const int cur = t & 1;

        if (t + 1 < T) {
            stage_tile((t + 1) * TILE_K, cur ^ 1);   // prefetch next tile into other buffer
            wait_async_le<NASY>();        // tile t landed; tile t+1 still in flight
        } else {
            wait_async_le<0>();
        }
        __syncthreads();                  // all waves' tile-t LDS writes visible

        // ---- 8 WMMA K-steps of 4 over tile t ----
        const float* Asb = As[cur];
        const float* Bsb = Bs[cur];
        #pragma unroll
        for (int kk = 0; kk < TILE_K; kk += 4) {
            const int ko = kk + khalf;
            v2f a[2], b[4];
            #pragma unroll
            for (int mt = 0; mt < 2; ++mt)
                a[mt] = *(const v2f*)&Asb[(waveM * 32 + mt * 16 + mlo) * LSTR + ko];
            #pragma unroll
            for (int nt = 0; nt < 4; ++nt)
                b[nt] = *(const v2f*)&Bsb[(waveN * 64 + nt * 16 + mlo) * LSTR + ko];
            #pragma unroll
            for (int mt = 0; mt < 2; ++mt)
                #pragma unroll
                for (int nt = 0; nt < 4; ++nt)
                    acc[mt][nt] = __builtin_amdgcn_wmma_f32_16x16x4_f32(
                        false, a[mt], false, b[nt], (short)0, acc[mt][nt], false, false);
        }
        __syncthreads();                  // buffer `cur` free for overwrite at t+2
    }

    // ---- epilogue + store ----
    float* Cz = C + (long)z * sCz;
    #pragma unroll
    for (int mt = 0; mt < 2; ++mt) {
        #pragma unroll
        for (int nt = 0; nt < 4; ++nt) {
            const int rbase = m0 + waveM * 32 + mt * 16 + ((lane >> 4) << 3);
            const int c     = n0 + waveN * 64 + nt * 16 + mlo;
            #pragma unroll
            for (int i = 0; i < 8; ++i) {
                const long r = rbase + i;
                float v = acc[mt][nt][i];
                if constexpr (EPI == 0) {
                    v += bias[c];
                } else if constexpr (EPI == 1) {
                    v = v * scale + add2d[(long)z * sAddz + r * (long)ldc + c];
                    if (mask[r * (long)ldc + c] == 0) v = -1e12f;
                } else if constexpr (EPI == 3) {
                    v += bias[c] + add2d[r * (long)ldc + c];
                }
                Cz[r * (long)ldc + c] = v;
            }
        }
    }
}

// One 256-thread block per attention row of length S=2048; in-place softmax.
__global__ __launch_bounds__(256)
void softmax_rows(float* __restrict__ attn, int S)
{
    __shared__ float red[256];
    const long row = blockIdx.x;
    float* p = attn + row * (long)S;
    const int tid = threadIdx.x;

    float v[8];
    float lmax = -3.402823466e38f;
    #pragma unroll
    for (int i = 0; i < 8; ++i) { v[i] = p[tid + (i << 8)]; lmax = fmaxf(lmax, v[i]); }
    red[tid] = lmax; __syncthreads();
    for (int s = 128; s > 0; s >>= 1) { if (tid < s) red[tid] = fmaxf(red[tid], red[tid + s]); __syncthreads(); }
    const float m = red[0];
    __syncthreads();

    float lsum = 0.f;
    #pragma unroll
    for (int i = 0; i < 8; ++i) { v[i] = __expf(v[i] - m); lsum += v[i]; }
    red[tid] = lsum; __syncthreads();
    for (int s = 128; s > 0; s >>= 1) { if (tid < s) red[tid] += red[tid + s]; __syncthreads(); }
    const float inv = 1.f / red[0];
    #pragma unroll
    for (int i = 0; i < 8; ++i) p[tid + (i << 8)] = v[i] * inv;
}

// One 256-thread block per hidden row of length H=1024; in-place LayerNorm (biased var, eps=1e-6).
__global__ __launch_bounds__(256)
void layernorm_rows(float* __restrict__ o, const float* __restrict__ g,
                    const float* __restrict__ b, int Hd)
{
    __shared__ float red[256];
    const long row = blockIdx.x;
    float* p = o + row * (long)Hd;
    const int tid = threadIdx.x;

    float v[4];
    float s = 0.f;
    #pragma unroll
    for (int i = 0; i < 4; ++i) { v[i] = p[tid + (i << 8)]; s += v[i]; }
    red[tid] = s; __syncthreads();
    for (int t = 128; t > 0; t >>= 1) { if (tid < t) red[tid] += red[tid + t]; __syncthreads(); }
    const float mu = red[0] * (1.f / 1024.f);
    __syncthreads();

    float s2 = 0.f;
    #pragma unroll
    for (int i = 0; i < 4; ++i) { float d = v[i] - mu; s2 += d * d; }
    red[tid] = s2; __syncthreads();
    for (int t = 128; t > 0; t >>= 1) { if (tid < t) red[tid] += red[tid + t]; __syncthreads(); }
    const float rstd = rsqrtf(red[0] * (1.f / 1024.f) + 1e-6f);
    #pragma unroll
    for (int i = 0; i < 4; ++i) {
        const int c = tid + (i << 8);
        p[c] = (v[i] - mu) * rstd * g[c] + b[c];
    }
}

extern "C" void kernel_launch(void* const* d_in, const int* in_sizes, int n_in,
                              void* d_out, int out_size, void* d_ws, size_t ws_size,
                              hipStream_t stream)
{
    constexpr int B = 8, S = 2048, H = 1024;
    constexpr long BS = (long)B * S;

    const float* x    = (const float*)d_in[0];
    const int*   mask = (const int*)  d_in[1];
    const float* ab   = (const float*)d_in[2];
    const float* Wq   = (const float*)d_in[3];
    const float* bq   = (const float*)d_in[4];
    const float* Wk   = (const float*)d_in[5];
    const float* bk   = (const float*)d_in[6];
    const float* Wv   = (const float*)d_in[7];
    const float* bv   = (const float*)d_in[8];
    const float* Wa   = (const float*)d_in[9];
    const float* ba   = (const float*)d_in[10];
    const float* gam  = (const float*)d_in[11];
    const float* bet  = (const float*)d_in[12];

    float* out  = (float*)d_out;                 // (B,S,H)
    float* attn = out + BS * H;                  // (B,S,S)

    float* q = (float*)d_ws;                     // (B*S, H)
    float* k = q + BS * H;
    float* v = k + BS * H;

    const float scale = 1.0f / (1.0f * 32.0f);   // 1/(TEMPERATURE*sqrt(H))
    const dim3 blk(256);

    // 1) Q/K/V projections: (16384 x 1024) * W^T + bias      [NT, EPI=bias]
    const dim3 gProj(H / TILE_N, (int)(BS / TILE_M), 1);
    gemm_wmma_f32<0, true><<<gProj, blk, 0, stream>>>(x, Wq, q, (int)BS, H, H, H, H, H,
                                                      bq, nullptr, nullptr, 1.f, 0, 0, 0, 0);
    gemm_wmma_f32<0, true><<<gProj, blk, 0, stream>>>(x, Wk, k, (int)BS, H, H, H, H, H,
                                                      bk, nullptr, nullptr, 1.f, 0, 0, 0, 0);
    gemm_wmma_f32<0, true><<<gProj, blk, 0, stream>>>(x, Wv, v, (int)BS, H, H, H, H, H,
                                                      bv, nullptr, nullptr, 1.f, 0, 0, 0, 0);

    // 2) scores = Q·K^T * scale + attn_bias, masked -> attn buffer   [NT, EPI=1, per batch]
    const dim3 gScore(S / TILE_N, S / TILE_M, B);
    gemm_wmma_f32<1, true><<<gScore, blk, 0, stream>>>(q, k, attn, S, S, H, H, H, S,
                                                       nullptr, ab, mask, scale,
                                                       (long)S * H, (long)S * H,
                                                       (long)S * S, (long)S * S);

    // 3) softmax rows, in place
    softmax_rows<<<dim3((unsigned)BS), blk, 0, stream>>>(attn, S);

    // 4) ctx = attn · V     [NN, EPI=plain, per batch]  (reuse q buffer)
    float* ctx = q;
    const dim3 gAV(H / TILE_N, S / TILE_M, B);
    gemm_wmma_f32<2, false><<<gAV, blk, 0, stream>>>(attn, v, ctx, S, H, S, S, H, H,
                                                     nullptr, nullptr, nullptr, 1.f,
                                                     (long)S * S, (long)S * H, (long)S * H, 0);

    // 5) out = ctx · Wa^T + ba + residual(x)   [NT, EPI=3]
    gemm_wmma_f32<3, true><<<gProj, blk, 0, stream>>>(ctx, Wa, out, (int)BS, H, H, H, H, H,
                                                      ba, x, nullptr, 1.f, 0, 0, 0, 0);

    // 6) LayerNorm rows of `out`, in place
    layernorm_rows<<<dim3((unsigned)BS), blk, 0, stream>>>(out, gam, bet, H);
}